// TransField_6390911336496
// MI455X (gfx1250) — compile-verified
//
#include <hip/hip_runtime.h>
#include <hip/hip_bf16.h>

#define CIN   8
#define COUT  16
#define RR    6
#define EPS_F 1e-12f

typedef __attribute__((ext_vector_type(2))) float v2f;
typedef __attribute__((ext_vector_type(8))) float v8f;

// ---------------------------------------------------------------------------
// Phase 0: zero contrib accumulators in workspace
// ---------------------------------------------------------------------------
__global__ void zero_kernel(float* __restrict__ p, size_t n) {
    size_t i = (size_t)blockIdx.x * blockDim.x + threadIdx.x;
    size_t stride = (size_t)gridDim.x * blockDim.x;
    for (; i < n; i += stride) p[i] = 0.0f;
}

// ---------------------------------------------------------------------------
// Phase 1: edge scatter.  One thread per edge.
//   contribAng_re[dst,c,r] -= xDiff[c] * lift_re[e,r,1]
//   contribAng_im[dst,c,r] -= xDiff[c] * lift_im[e,r,1]
//   contribMag  [dst,c,r] += x[src,c] * sqrt(re0^2 + im0^2 + eps)
// Streaming data (lift, edges) uses nontemporal loads; x gathers stay cached.
// ---------------------------------------------------------------------------
__global__ __launch_bounds__(256) void edge_kernel(
    const float* __restrict__ x,
    const int*   __restrict__ edges,
    const float* __restrict__ lre,
    const float* __restrict__ lim,
    float* __restrict__ ang_re,
    float* __restrict__ ang_im,
    float* __restrict__ mag,
    int n_edges)
{
    int e = blockIdx.x * blockDim.x + threadIdx.x;
    if (e >= n_edges) return;

    int src = __builtin_nontemporal_load(edges + 2 * (size_t)e + 0);
    int dst = __builtin_nontemporal_load(edges + 2 * (size_t)e + 1);

    // gather x[src], x[dst] (8 floats each, 32B aligned rows)
    const float4* xs4 = reinterpret_cast<const float4*>(x + (size_t)src * CIN);
    const float4* xd4 = reinterpret_cast<const float4*>(x + (size_t)dst * CIN);
    float4 s0 = xs4[0], s1 = xs4[1];
    float4 d0 = xd4[0], d1 = xd4[1];
    float xsrc[CIN] = { s0.x, s0.y, s0.z, s0.w, s1.x, s1.y, s1.z, s1.w };
    float xdif[CIN] = { s0.x - d0.x, s0.y - d0.y, s0.z - d0.z, s0.w - d0.w,
                        s1.x - d1.x, s1.y - d1.y, s1.z - d1.z, s1.w - d1.w };

    const float* lr = lre + (size_t)e * RR * 2;
    const float* li = lim + (size_t)e * RR * 2;
    float* ar = ang_re + (size_t)dst * CIN * RR;
    float* ai = ang_im + (size_t)dst * CIN * RR;
    float* am = mag    + (size_t)dst * CIN * RR;

#pragma unroll
    for (int r = 0; r < RR; ++r) {
        float re0 = __builtin_nontemporal_load(lr + 2 * r + 0);
        float re1 = __builtin_nontemporal_load(lr + 2 * r + 1);
        float im0 = __builtin_nontemporal_load(li + 2 * r + 0);
        float im1 = __builtin_nontemporal_load(li + 2 * r + 1);
        float l0  = sqrtf(fmaf(re0, re0, fmaf(im0, im0, EPS_F)));
#pragma unroll
        for (int c = 0; c < CIN; ++c) {
            atomicAdd(ar + c * RR + r, -(xdif[c] * re1));
            atomicAdd(ai + c * RR + r, -(xdif[c] * im1));
            atomicAdd(am + c * RR + r, xsrc[c] * l0);
        }
    }
}

// ---------------------------------------------------------------------------
// Phase 2: per-node dense stage with V_WMMA_F32_16X16X4_F32.
// One wave handles one 16-node tile. For each input channel c:
//   sA_re/sA_im/sM (16x16, node x cout) = contrib(16 x K=6) * zonal(K=6 x 16)
// via two chained K=4 WMMAs (second chunk zero-padded K=6,7).
// Then per-element: phi = atan2(im,re)+phase[o,c]; rho = sqrt(m^2+eps);
// out += rho * (cos(phi) + i sin(phi)), summed over c.
// ---------------------------------------------------------------------------
__global__ __launch_bounds__(256) void node_kernel(
    const float* __restrict__ ang_re,
    const float* __restrict__ ang_im,
    const float* __restrict__ mag,
    const float* __restrict__ zonalAng,
    const float* __restrict__ zonalMag,
    const float* __restrict__ phase,
    float* __restrict__ out,     // (n_nodes, COUT) complex64, interleaved
    int n_nodes)
{
    const int lane   = threadIdx.x & 31;
    const int wave   = threadIdx.x >> 5;
    const int tile   = blockIdx.x * (blockDim.x >> 5) + wave;
    const int node0  = tile * 16;
    if (node0 >= n_nodes) return;          // wave-uniform exit

    const bool laneLo = (lane < 16);
    const int  m = lane & 15;              // A-matrix row (node within tile)
    const int  o = lane & 15;              // B/C/D column (output channel)

    v8f out_re = {};
    v8f out_im = {};

    for (int c = 0; c < CIN; ++c) {
        // ---- B operands: zonal[o,c,r] -> B[k,o] ----
        // chunk0: lanes 0-15 hold rows K=0,1 (r=0,1); lanes 16-31 rows K=2,3 (r=2,3)
        const float* zA = zonalAng + ((size_t)o * CIN + c) * RR;
        const float* zM = zonalMag + ((size_t)o * CIN + c) * RR;
        const int rb = laneLo ? 0 : 2;
        v2f bA0 = { zA[rb], zA[rb + 1] };
        v2f bM0 = { zM[rb], zM[rb + 1] };
        // chunk1: rows K=0,1 -> r=4,5 (lanes 0-15); rows K=2,3 -> zero pad
        v2f bA1 = { 0.0f, 0.0f };
        v2f bM1 = { 0.0f, 0.0f };
        if (laneLo) { bA1[0] = zA[4]; bA1[1] = zA[5];
                      bM1[0] = zM[4]; bM1[1] = zM[5]; }

        // ---- A operands: contrib[node0+m, c, r] ----
        const size_t ab = ((size_t)(node0 + m) * CIN + c) * RR;
        const int ra = laneLo ? 0 : 2;
        v2f aRe0 = { ang_re[ab + ra], ang_re[ab + ra + 1] };
        v2f aIm0 = { ang_im[ab + ra], ang_im[ab + ra + 1] };
        v2f aMg0 = { mag   [ab + ra], mag   [ab + ra + 1] };
        v2f aRe1 = { 0.0f, 0.0f };
        v2f aIm1 = { 0.0f, 0.0f };
        v2f aMg1 = { 0.0f, 0.0f };
        if (laneLo) { aRe1[0] = ang_re[ab + 4]; aRe1[1] = ang_re[ab + 5];
                      aIm1[0] = ang_im[ab + 4]; aIm1[1] = ang_im[ab + 5];
                      aMg1[0] = mag   [ab + 4]; aMg1[1] = mag   [ab + 5]; }

        // ---- three 16x16xK=6 products, each as two chained K=4 WMMAs ----
        v8f sRe = {};
        sRe = __builtin_amdgcn_wmma_f32_16x16x4_f32(false, aRe0, false, bA0,
                                                    (short)0, sRe, false, false);
        sRe = __builtin_amdgcn_wmma_f32_16x16x4_f32(false, aRe1, false, bA1,
                                                    (short)0, sRe, false, false);
        v8f sIm = {};
        sIm = __builtin_amdgcn_wmma_f32_16x16x4_f32(false, aIm0, false, bA0,
                                                    (short)0, sIm, false, false);
        sIm = __builtin_amdgcn_wmma_f32_16x16x4_f32(false, aIm1, false, bA1,
                                                    (short)0, sIm, false, false);
        v8f sMg = {};
        sMg = __builtin_amdgcn_wmma_f32_16x16x4_f32(false, aMg0, false, bM0,
                                                    (short)0, sMg, false, false);
        sMg = __builtin_amdgcn_wmma_f32_16x16x4_f32(false, aMg1, false, bM1,
                                                    (short)0, sMg, false, false);

        // ---- transcendental epilogue, summed over c ----
        const float ph = phase[o * CIN + c];
#pragma unroll
        for (int j = 0; j < 8; ++j) {
            float re  = sRe[j];
            float im  = sIm[j];
            float mm  = sMg[j];
            float phi = atan2f(im, re) + ph;
            float rho = sqrtf(fmaf(mm, mm, EPS_F));
            float sn, cs;
            __sincosf(phi, &sn, &cs);
            out_re[j] += rho * cs;
            out_im[j] += rho * sn;
        }
    }

    // ---- write out: C/D layout: VGPR j -> node row j (lanes 0-15) / j+8 ----
    const int moff = laneLo ? 0 : 8;
#pragma unroll
    for (int j = 0; j < 8; ++j) {
        int node = node0 + j + moff;
        size_t idx = ((size_t)node * COUT + o) * 2;
        out[idx + 0] = out_re[j];
        out[idx + 1] = out_im[j];
    }
}

// ---------------------------------------------------------------------------
extern "C" void kernel_launch(void* const* d_in, const int* in_sizes, int n_in,
                              void* d_out, int out_size, void* d_ws, size_t ws_size,
                              hipStream_t stream) {
    const float* x     = (const float*)d_in[0];
    const int*   edges = (const int*)  d_in[1];
    const float* lre   = (const float*)d_in[2];
    const float* lim   = (const float*)d_in[3];
    const float* zA    = (const float*)d_in[4];
    const float* zM    = (const float*)d_in[5];
    const float* ph    = (const float*)d_in[6];

    const int n_nodes = in_sizes[0] / CIN;
    const int n_edges = in_sizes[1] / 2;

    float* ws = (float*)d_ws;
    const size_t per = (size_t)n_nodes * CIN * RR;
    float* ang_re = ws;
    float* ang_im = ws + per;
    float* mg     = ws + 2 * per;

    zero_kernel<<<2048, 256, 0, stream>>>(ws, 3 * per);

    edge_kernel<<<(n_edges + 255) / 256, 256, 0, stream>>>(
        x, edges, lre, lim, ang_re, ang_im, mg, n_edges);

    const int tiles = (n_nodes + 15) / 16;
    const int wavesPerBlk = 8;  // 256 threads
    node_kernel<<<(tiles + wavesPerBlk - 1) / wavesPerBlk, wavesPerBlk * 32, 0, stream>>>(
        ang_re, ang_im, mg, zA, zM, ph, (float*)d_out, n_nodes);
}